// CompConv_10290741641561
// MI455X (gfx1250) — compile-verified
//
#include <hip/hip_runtime.h>

typedef __attribute__((ext_vector_type(2))) float v2f;
typedef __attribute__((ext_vector_type(8))) float v8f;

#define D_DIM 64
// W stored k-paired in LDS: Wp[kp*KP_STRIDE + n*2 + (k&1)], kp = k>>1 (32 pairs).
// KP_STRIDE = 160 floats: 160 mod 64 = 32 banks, so consecutive kp (read by the
// two lane halves) occupy disjoint bank halves -> conflict-free ds_load_b64.
#define KP_STRIDE 160

// ---------------------------------------------------------------------------
// Kernel 1: zero the accumulator (d_out doubles as agg) and the degree array.
// ---------------------------------------------------------------------------
__global__ __launch_bounds__(256) void compconv_zero_kernel(float* __restrict__ agg, int n_agg,
                                                            float* __restrict__ deg, int n_deg) {
    int i = blockIdx.x * 256 + threadIdx.x;
    if (i < n_agg) agg[i] = 0.0f;
    if (i < n_deg) deg[i] = 0.0f;
}

// ---------------------------------------------------------------------------
// Kernel 2: edge scatter. 16 threads per edge, each owns one float4 chunk of
// the 64-wide feature row. feat + agg are L2-resident (12.8 MB each).
// ---------------------------------------------------------------------------
__global__ __launch_bounds__(256) void compconv_scatter_kernel(const float4* __restrict__ feat4,
                                                               const int* __restrict__ src,
                                                               const int* __restrict__ dst,
                                                               float* __restrict__ agg,
                                                               float* __restrict__ deg,
                                                               int n_edges) {
    int gid = blockIdx.x * 256 + threadIdx.x;
    int e = gid >> 4;
    int c = gid & 15;
    if (e >= n_edges) return;
    int s = src[e];
    int d = dst[e];
    float4 v = feat4[(size_t)s * 16 + c];
    float* p = agg + (size_t)d * D_DIM + c * 4;
    atomicAdd(p + 0, v.x);
    atomicAdd(p + 1, v.y);
    atomicAdd(p + 2, v.z);
    atomicAdd(p + 3, v.w);
    if (c == 0) atomicAdd(deg + d, 1.0f);
}

// ---------------------------------------------------------------------------
// Kernel 3: out = ((agg - deg*h_e)/max(deg,1)) @ W via V_WMMA_F32_16X16X4_F32.
// One wave per 16-row tile; 4 column tiles x 16 k-steps = 64 WMMAs/wave.
// A-row loads batched up front; B operands are single aligned ds_load_b64
// from the k-paired LDS copy of W. In-place on `data`: a wave reads only its
// own 16 rows before writing them.
// ---------------------------------------------------------------------------
__global__ __launch_bounds__(256) void compconv_gemm_kernel(const float* __restrict__ he,
                                                            const float* __restrict__ W,
                                                            const float* __restrict__ deg,
                                                            float* data,   // agg in, out out
                                                            int n_nodes) {
    __shared__ float Wlds[32 * KP_STRIDE];   // 20 KB
    __shared__ float He[D_DIM];

    // Stage W (k-paired, bank-staggered) and h_e into LDS.
    for (int i = threadIdx.x; i < D_DIM * D_DIM; i += 256) {
        int r = i >> 6, c = i & 63;
        Wlds[(r >> 1) * KP_STRIDE + c * 2 + (r & 1)] = W[i];
    }
    if (threadIdx.x < D_DIM) He[threadIdx.x] = he[threadIdx.x];
    __syncthreads();

    const int wave = threadIdx.x >> 5;
    const int lane = threadIdx.x & 31;
    const int tile = blockIdx.x * 8 + wave;
    const int m0 = tile * 16;
    if (m0 >= n_nodes) return;           // wave-uniform: EXEC stays all-1s

    const int half = lane >> 4;          // 0: k,k+1   1: k+2,k+3
    const int l15  = lane & 15;
    int m = m0 + l15;
    if (m > n_nodes - 1) m = n_nodes - 1;  // clamp for partial tiles (reads only)

    const float degm = deg[m];
    const float invd = 1.0f / fmaxf(degm, 1.0f);
    const float* arow = data + (size_t)m * D_DIM + half * 2;

    // Batch all 16 A-operand loads (b64 each) so they pipeline with one wait.
    v2f araw[16];
#pragma unroll
    for (int ks = 0; ks < 16; ++ks) {
        araw[ks] = *(const v2f*)(arow + ks * 4);
    }

    v8f acc0 = {}, acc1 = {}, acc2 = {}, acc3 = {};

#pragma unroll
    for (int ks = 0; ks < 16; ++ks) {
        const int koff = ks * 4 + half * 2;
        // A operand with h_e / degree normalization folded in.
        v2f h2 = *(const v2f*)&He[koff];                 // broadcast ds_load_b64
        v2f a = (araw[ks] - degm * h2) * invd;

        const float* wp = &Wlds[(ks * 2 + half) * KP_STRIDE + l15 * 2];
        v2f b0 = *(const v2f*)(wp + 0);                  // col tile 0
        v2f b1 = *(const v2f*)(wp + 32);                 // col tile 1
        v2f b2 = *(const v2f*)(wp + 64);                 // col tile 2
        v2f b3 = *(const v2f*)(wp + 96);                 // col tile 3

        acc0 = __builtin_amdgcn_wmma_f32_16x16x4_f32(false, a, false, b0, (short)0, acc0, false, false);
        acc1 = __builtin_amdgcn_wmma_f32_16x16x4_f32(false, a, false, b1, (short)0, acc1, false, false);
        acc2 = __builtin_amdgcn_wmma_f32_16x16x4_f32(false, a, false, b2, (short)0, acc2, false, false);
        acc3 = __builtin_amdgcn_wmma_f32_16x16x4_f32(false, a, false, b3, (short)0, acc3, false, false);
    }

    // C/D layout: VGPR j -> row m0+j (lanes 0-15) / m0+8+j (lanes 16-31), col = lane&15.
#pragma unroll
    for (int j = 0; j < 8; ++j) {
        int row = m0 + j + half * 8;
        if (row < n_nodes) {
            float* orow = data + (size_t)row * D_DIM + l15;
            orow[ 0] = acc0[j];
            orow[16] = acc1[j];
            orow[32] = acc2[j];
            orow[48] = acc3[j];
        }
    }
}

// ---------------------------------------------------------------------------
// Inputs: feat (N*64 f32), h_e (64 f32), W (64*64 f32), src (E i32), dst (E i32)
// Output: (N*64 f32). Workspace: deg (N f32).
// ---------------------------------------------------------------------------
extern "C" void kernel_launch(void* const* d_in, const int* in_sizes, int n_in,
                              void* d_out, int out_size, void* d_ws, size_t ws_size,
                              hipStream_t stream) {
    const float* feat = (const float*)d_in[0];
    const float* he   = (const float*)d_in[1];
    const float* W    = (const float*)d_in[2];
    const int*   src  = (const int*)d_in[3];
    const int*   dst  = (const int*)d_in[4];
    float* out = (float*)d_out;
    float* deg = (float*)d_ws;

    const int n_nodes = in_sizes[0] / D_DIM;
    const int n_edges = in_sizes[3];

    // 1) zero agg (== d_out) and deg
    const int n_agg = n_nodes * D_DIM;
    compconv_zero_kernel<<<(n_agg + 255) / 256, 256, 0, stream>>>(out, n_agg, deg, n_nodes);

    // 2) scatter raw features + degrees
    const int scatter_threads = n_edges * 16;
    compconv_scatter_kernel<<<(scatter_threads + 255) / 256, 256, 0, stream>>>(
        (const float4*)feat, src, dst, out, deg, n_edges);

    // 3) fused normalize + GEMM via WMMA f32 16x16x4
    const int tiles = (n_nodes + 15) / 16;
    compconv_gemm_kernel<<<(tiles + 7) / 8, 256, 0, stream>>>(he, W, deg, out, n_nodes);
}